// GCNWeightedLayer_32435593019463
// MI455X (gfx1250) — compile-verified
//
#include <hip/hip_runtime.h>
#include <math.h>

typedef __attribute__((ext_vector_type(2))) float v2f;
typedef __attribute__((ext_vector_type(8))) float v8f;

#define NN   8192
#define FF   128
#define EPSF 1e-8f
#define NSEG 4              // K-split factor for stage-1 parallelism

// ---------------------------------------------------------------------------
// Kernel 1: degree[i] = sum_j (adj+la+eps*noise)[i,j] + 1 ; dinv = (deg+eps)^-1/2
// One block per row; float4 (global_load_b128) streaming of the 3 matrices.
// ---------------------------------------------------------------------------
__global__ __launch_bounds__(256) void k_degree(const float* __restrict__ adj,
                                                const float* __restrict__ la,
                                                const float* __restrict__ noise,
                                                float* __restrict__ dinv) {
    __shared__ float red[256];
    const int row = blockIdx.x;
    const size_t base4 = (size_t)row * (NN / 4);
    const float4* __restrict__ a4 = (const float4*)adj;
    const float4* __restrict__ l4 = (const float4*)la;
    const float4* __restrict__ n4 = (const float4*)noise;
    float p = 0.0f;
#pragma unroll 2
    for (int j = threadIdx.x; j < NN / 4; j += 256) {
        float4 a = a4[base4 + j];
        float4 l = l4[base4 + j];
        float4 n = n4[base4 + j];
        p += (a.x + l.x + EPSF * n.x) + (a.y + l.y + EPSF * n.y) +
             (a.z + l.z + EPSF * n.z) + (a.w + l.w + EPSF * n.w);
    }
    red[threadIdx.x] = p;
    __syncthreads();
    for (int s = 128; s > 0; s >>= 1) {
        if (threadIdx.x < s) red[threadIdx.x] += red[threadIdx.x + s];
        __syncthreads();
    }
    if (threadIdx.x == 0) {
        float deg = red[0] + 1.0f;                 // self loop (eye)
        dinv[row] = 1.0f / sqrtf(deg + EPSF);
    }
}

// ---------------------------------------------------------------------------
// Kernel 2: s2[j,f] = dinv[j] * sum_k x[j,k] * W[k,f]   (tiny GEMM, 0.27 GFLOP)
// ---------------------------------------------------------------------------
__global__ __launch_bounds__(128) void k_support(const float* __restrict__ x,
                                                 const float* __restrict__ w,
                                                 const float* __restrict__ dinv,
                                                 float* __restrict__ s2) {
    __shared__ float xr[FF];
    const int row = blockIdx.x;
    const int f = threadIdx.x;
    xr[f] = x[(size_t)row * FF + f];
    __syncthreads();
    float acc = 0.0f;
#pragma unroll 8
    for (int k = 0; k < FF; ++k) {
        acc += xr[k] * w[k * FF + f];
    }
    s2[(size_t)row * FF + f] = acc * dinv[row];
}

// ---------------------------------------------------------------------------
// Kernel 3 (stage 1): partial[s][i,f] = sum_{j in seg s} A[i,j] * s2[j,f]
// A recomputed on the fly: adj + la + eps*noise (+1 on diagonal, lands in the
// segment containing the row automatically).
// Block = 8 waves (256 threads), 128 rows, full 128 cols, K chunked by 32.
// grid = (M-blocks, NSEG). If fuseScale!=0 (single-segment fallback), the
// epilogue applies dinv[i] and writes d_out directly.
// ---------------------------------------------------------------------------
#define KC  32             // K-chunk
#define MB  128            // rows per block
#define APITCH 34          // dwords: conflict-free, 8B-aligned A fragments
#define BPITCH (2*FF + 32) // 288 dwords: halves' bank spans disjoint (288%64==32)

__global__ __launch_bounds__(256) void k_spmm(const float* __restrict__ adj,
                                              const float* __restrict__ la,
                                              const float* __restrict__ noise,
                                              const float* __restrict__ s2,
                                              const float* __restrict__ dinv,
                                              float* __restrict__ dst,
                                              int segLen, int fuseScale) {
    __shared__ float As[MB][APITCH];        // 128 x 34 dwords (~17.4 KB)
    __shared__ float Bs[KC / 2][BPITCH];    // 16 x 288 dwords (~18.4 KB)

    const int t    = threadIdx.x;
    const int wave = t >> 5;
    const int lane = t & 31;
    const int half = lane >> 4;   // K-half selector of the fragment
    const int l16  = lane & 15;   // M (for A) / N (for B/C) index
    const int rowBase = blockIdx.x * MB;
    const int segBase = blockIdx.y * segLen;
    const int segEnd  = segBase + segLen;

    v8f acc[8];
#pragma unroll
    for (int i = 0; i < 8; ++i)
#pragma unroll
        for (int e = 0; e < 8; ++e) acc[i][e] = 0.0f;

    for (int k0 = segBase; k0 < segEnd; k0 += KC) {
        // ---- stage A chunk (MB x KC), coalesced 128B rows per wave ----
#pragma unroll
        for (int e = 0; e < (MB * KC) / 256; ++e) {   // 16 iters
            int idx = e * 256 + t;
            int r = idx >> 5;          // 0..127
            int c = idx & (KC - 1);    // 0..31
            int gr = rowBase + r;
            int gc = k0 + c;
            size_t g = (size_t)gr * NN + gc;
            float a = adj[g] + la[g] + EPSF * noise[g];
            if (gr == gc) a += 1.0f;   // self loop
            As[r][c] = a;
        }
        // ---- stage B chunk (KC x FF) K-pair interleaved: Bs[r/2][c*2+(r&1)] ----
#pragma unroll
        for (int e = 0; e < (KC * FF) / 256; ++e) {   // 16 iters
            int idx = e * 256 + t;
            int r = idx >> 7;          // 0..31
            int c = idx & (FF - 1);    // 0..127
            Bs[r >> 1][c * 2 + (r & 1)] = s2[(size_t)(k0 + r) * FF + c];
        }
        // ---- prefetch next A chunk (1 line covers a full 32-float K-row) ----
        if (k0 + KC < segEnd) {
#pragma unroll
            for (int e = 0; e < 2; ++e) {
                int idx = e * 256 + t;          // 0..511, need 384
                if (idx < 3 * MB) {
                    int m = idx >> 7;           // which matrix
                    int r = idx & (MB - 1);
                    const float* src = (m == 0) ? adj : (m == 1) ? la : noise;
                    __builtin_prefetch(&src[(size_t)(rowBase + r) * NN + k0 + KC], 0, 0);
                }
            }
        }
        __syncthreads();

        // ---- WMMA core: 8 k-steps x 8 N-tiles = 64 wmma per chunk ----
#pragma unroll
        for (int kk = 0; kk < KC; kk += 4) {
            const int kf = kk + 2 * half;              // even -> 8B aligned
            const int kp = (kk >> 1) + half;           // interleaved B row
            v2f a = *(const v2f*)&As[wave * 16 + l16][kf];
#pragma unroll
            for (int nt = 0; nt < 8; ++nt) {
                v2f b = *(const v2f*)&Bs[kp][(nt * 16 + l16) * 2];
                acc[nt] = __builtin_amdgcn_wmma_f32_16x16x4_f32(
                    /*neg_a=*/false, a, /*neg_b=*/false, b,
                    /*c_mod=*/(short)0, acc[nt],
                    /*reuse_a=*/false, /*reuse_b=*/false);
            }
        }
        __syncthreads();
    }

    // ---- epilogue ----
    // C layout: VGPR v, lane L -> M = v + 8*(L/16), N = L%16
    float* segDst = dst + (size_t)blockIdx.y * NN * FF;
    if (fuseScale) {
        float dv[8];
#pragma unroll
        for (int v = 0; v < 8; ++v)
            dv[v] = dinv[rowBase + wave * 16 + v + 8 * half];
#pragma unroll
        for (int nt = 0; nt < 8; ++nt)
#pragma unroll
            for (int v = 0; v < 8; ++v) {
                int gRow = rowBase + wave * 16 + v + 8 * half;
                segDst[(size_t)gRow * FF + nt * 16 + l16] = acc[nt][v] * dv[v];
            }
    } else {
#pragma unroll
        for (int nt = 0; nt < 8; ++nt)
#pragma unroll
            for (int v = 0; v < 8; ++v) {
                int gRow = rowBase + wave * 16 + v + 8 * half;
                segDst[(size_t)gRow * FF + nt * 16 + l16] = acc[nt][v];
            }
    }
}

// ---------------------------------------------------------------------------
// Kernel 4 (stage 2): out[i,f] = dinv[i] * sum_s partial[s][i,f]   (float4)
// ---------------------------------------------------------------------------
__global__ __launch_bounds__(256) void k_reduce(const float* __restrict__ partial,
                                                const float* __restrict__ dinv,
                                                float* __restrict__ out) {
    const int o = blockIdx.x * 256 + threadIdx.x;   // float4 index, NN*FF/4 total
    const float4* p4 = (const float4*)partial;
    float4 s = p4[o];
#pragma unroll
    for (int seg = 1; seg < NSEG; ++seg) {
        float4 q = p4[o + (size_t)seg * (NN * FF / 4)];
        s.x += q.x; s.y += q.y; s.z += q.z; s.w += q.w;
    }
    const float d = dinv[o >> (7 - 2)];             // row = (o*4)/128
    float4 r; r.x = s.x * d; r.y = s.y * d; r.z = s.z * d; r.w = s.w * d;
    ((float4*)out)[o] = r;
}

// ---------------------------------------------------------------------------
extern "C" void kernel_launch(void* const* d_in, const int* in_sizes, int n_in,
                              void* d_out, int out_size, void* d_ws, size_t ws_size,
                              hipStream_t stream) {
    const float* x     = (const float*)d_in[0];   // (N, 128)
    const float* adj   = (const float*)d_in[1];   // (N, N)
    const float* noise = (const float*)d_in[2];   // (N, N)
    const float* w     = (const float*)d_in[3];   // (128, 128)
    const float* la    = (const float*)d_in[4];   // (N, N)
    float* out = (float*)d_out;

    // workspace layout: dinv (32KB) | s2 (4MB) | partial (NSEG * 4MB, optional)
    float* dinv    = (float*)d_ws;
    float* s2      = dinv + NN;
    float* partial = s2 + (size_t)NN * FF;
    const size_t needSplit = ((size_t)NN + (size_t)NN * FF * (1 + NSEG)) * sizeof(float);

    k_degree <<<NN, 256, 0, stream>>>(adj, la, noise, dinv);
    k_support<<<NN, 128, 0, stream>>>(x, w, dinv, s2);

    if (ws_size >= needSplit) {
        // K-split: 256 blocks / 2048 waves, deterministic 2-stage reduce.
        k_spmm  <<<dim3(NN / MB, NSEG), 256, 0, stream>>>(adj, la, noise, s2, dinv,
                                                          partial, NN / NSEG, 0);
        k_reduce<<<NN * FF / 4 / 256, 256, 0, stream>>>(partial, dinv, out);
    } else {
        // Fallback: single segment, fused dinv scale, direct store to out.
        k_spmm  <<<dim3(NN / MB, 1), 256, 0, stream>>>(adj, la, noise, s2, dinv,
                                                       out, NN, 1);
    }
}